// enmf_frame_84731114816066
// MI455X (gfx1250) — compile-verified
//
#include <hip/hip_runtime.h>
#include <hip/hip_bf16.h>
#include <math.h>

// ---------------------------------------------------------------------------
// ENMF loss on MI455X (gfx1250).
// Dominant cost is streaming item_emb (256 MB) for the 64x64 Gram and the
// 210 MB gathered history rows -> memory bound (~20-25us at 23.3 TB/s).
// The Gram uses V_WMMA_F32_16X16X4_F32: one wave owns the full 64x64 f32
// accumulator (16 tiles x 8 VGPRs) and the A/B fragments for a 4-row chunk
// coincide (mirrored 16x4 / 4x16 layouts), so 4 v2f fragments feed all 16
// WMMAs per chunk. Kernel is templated on GATHER so the indexed (user-batch)
// variant gets a branch-free inner loop instead of a per-iteration null test.
// ---------------------------------------------------------------------------

typedef __attribute__((ext_vector_type(2))) float v2f;
typedef __attribute__((ext_vector_type(8))) float v8f;

#define DIM 64

// ws layout (floats)
#define WS_GITEM 0        // 64*64
#define WS_GUSER 4096     // 64*64
#define WS_POS   8192     // scalar
#define WS_UNORM 8193     // scalar
#define WS_N     8194

__global__ void enmf_zero_kernel(float* ws, int n) {
    int i = blockIdx.x * blockDim.x + threadIdx.x;
    if (i < n) ws[i] = 0.0f;
}

// Gram = sum over rows of emb[row]^T emb[row]  (64x64), rows optionally
// gathered through idx (GATHER=true). One wave accumulates the full 64x64 in
// registers, grid-strides over 4-row chunks, then atomically reduces.
template <bool GATHER>
__global__ void __launch_bounds__(128)
enmf_gram_wmma_kernel(const float* __restrict__ emb,
                      const int* __restrict__ idx,
                      int nrows, float* __restrict__ gram_out) {
    const int lane  = threadIdx.x & 31;
    const int wvloc = threadIdx.x >> 5;
    const int wave  = blockIdx.x * (blockDim.x >> 5) + wvloc;
    const int nwav  = gridDim.x * (blockDim.x >> 5);
    const int half  = lane >> 4;     // lanes 0-15 -> K=0,1 ; lanes 16-31 -> K=2,3
    const int col   = lane & 15;
    const int k0    = half * 2;

    v8f zero = {};
    v8f acc[4][4];
#pragma unroll
    for (int i = 0; i < 4; ++i)
#pragma unroll
        for (int j = 0; j < 4; ++j) acc[i][j] = zero;

    const int nchunks = nrows >> 2;     // 4 source rows per chunk (K=4)
    for (int c = wave; c < nchunks; c += nwav) {
        const int r0 = c << 2;
        long long rowA, rowB;
        if (GATHER) {
            rowA = (long long)idx[r0 + k0];
            rowB = (long long)idx[r0 + k0 + 1];
        } else {
            rowA = (long long)(r0 + k0);
            rowB = (long long)(r0 + k0 + 1);
        }
        const float* pA = emb + rowA * DIM + col;
        const float* pB = emb + rowB * DIM + col;
        v2f f[4];
#pragma unroll
        for (int i = 0; i < 4; ++i) {   // 4 d-blocks of 16 columns
            f[i].x = pA[i * 16];        // K = k0     row
            f[i].y = pB[i * 16];        // K = k0 + 1 row
        }
#pragma unroll
        for (int i = 0; i < 4; ++i)
#pragma unroll
            for (int j = 0; j < 4; ++j)
                acc[i][j] = __builtin_amdgcn_wmma_f32_16x16x4_f32(
                    false, f[i], false, f[j], (short)0, acc[i][j], false, false);
    }

    // C/D layout: VGPR r, lanes 0-15 -> M=r, lanes 16-31 -> M=r+8; N=lane%16.
#pragma unroll
    for (int i = 0; i < 4; ++i)
#pragma unroll
        for (int j = 0; j < 4; ++j)
#pragma unroll
            for (int r = 0; r < 8; ++r) {
                const int M = i * 16 + r + 8 * half;
                const int N = j * 16 + col;
                atomicAdd(&gram_out[M * DIM + N], acc[i][j][r]);
            }
}

// pos part: sum over (b,l) of 0.5*p^2 - 2*p, p = <u[b]*h, item[hist[u_b,l]]>.
// One block per batch element; each wave owns 2 dims/lane of w=u*h and
// butterfly-reduces a 64-float dot per history item.
__global__ void __launch_bounds__(256)
enmf_pos_kernel(const int* __restrict__ users, const int* __restrict__ hist,
                const float* __restrict__ user_emb,
                const float* __restrict__ item_emb,
                const float* __restrict__ Hw, int hist_len,
                float* __restrict__ pos_acc) {
    const int b    = blockIdx.x;
    const int lane = threadIdx.x & 31;
    const int wave = threadIdx.x >> 5;
    const int nw   = blockDim.x >> 5;
    const int uid  = users[b];
    const int d0   = lane * 2;

    const float w0 = user_emb[(long long)uid * DIM + d0]     * Hw[d0];
    const float w1 = user_emb[(long long)uid * DIM + d0 + 1] * Hw[d0 + 1];
    const int* hrow = hist + (long long)uid * hist_len;

    float local = 0.0f;
    for (int l = wave; l < hist_len; l += nw) {
        const int it = hrow[l];
        const float2 v = *(const float2*)(item_emb + (long long)it * DIM + d0);
        float p = v.x * w0 + v.y * w1;
#pragma unroll
        for (int m = 16; m > 0; m >>= 1) p += __shfl_xor(p, m, 32);
        if (lane == 0) local += 0.5f * p * p - 2.0f * p;
    }
    if (lane == 0) atomicAdd(pos_acc, local);
}

// sum of squares (float4 streamed), n divisible by 4
__global__ void __launch_bounds__(256)
enmf_sumsq_kernel(const float* __restrict__ x, long long n4,
                  float* __restrict__ out) {
    long long i = (long long)blockIdx.x * blockDim.x + threadIdx.x;
    const long long stride = (long long)gridDim.x * blockDim.x;
    float s = 0.0f;
    for (long long k = i; k < n4; k += stride) {
        const float4 v = ((const float4*)x)[k];
        s += v.x * v.x + v.y * v.y + v.z * v.z + v.w * v.w;
    }
#pragma unroll
    for (int m = 16; m > 0; m >>= 1) s += __shfl_xor(s, m, 32);
    if ((threadIdx.x & 31) == 0) atomicAdd(out, s);
}

// epilogue: t = sum(G_item .* G_user .* h h^T); item norm^2 = trace(G_item)
__global__ void __launch_bounds__(64)
enmf_final_kernel(const float* __restrict__ G_item,
                  const float* __restrict__ G_user,
                  const float* __restrict__ Hw,
                  const float* __restrict__ pos_acc,
                  const float* __restrict__ unorm2,
                  float* __restrict__ out) {
    __shared__ float st[DIM], str[DIM];
    const int d1 = threadIdx.x;
    float rowt = 0.0f;
    for (int d2 = 0; d2 < DIM; ++d2)
        rowt += G_item[d1 * DIM + d2] * G_user[d1 * DIM + d2] * Hw[d2];
    st[d1]  = Hw[d1] * rowt;
    str[d1] = G_item[d1 * (DIM + 1)];   // diagonal -> trace
    __syncthreads();
    if (d1 == 0) {
        float t = 0.0f, tr = 0.0f;
        for (int k = 0; k < DIM; ++k) { t += st[k]; tr += str[k]; }
        const float reg  = 1e-4f * (sqrtf(*unorm2) + sqrtf(tr));
        const float loss = 0.5f * t + *pos_acc + reg;
        out[0] = loss; out[1] = reg; out[2] = reg;
    }
}

extern "C" void kernel_launch(void* const* d_in, const int* in_sizes, int n_in,
                              void* d_out, int out_size, void* d_ws, size_t ws_size,
                              hipStream_t stream) {
    const int*   users    = (const int*)d_in[0];
    const int*   hist     = (const int*)d_in[1];
    const float* user_emb = (const float*)d_in[2];
    const float* item_emb = (const float*)d_in[3];
    const float* Hw       = (const float*)d_in[4];

    const int batch    = in_sizes[0];
    const int n_users  = in_sizes[2] / DIM;
    const int n_items  = in_sizes[3] / DIM;
    const int hist_len = in_sizes[1] / n_users;

    float* ws      = (float*)d_ws;
    float* G_item  = ws + WS_GITEM;
    float* G_user  = ws + WS_GUSER;
    float* pos_acc = ws + WS_POS;
    float* unorm2  = ws + WS_UNORM;

    enmf_zero_kernel<<<(WS_N + 255) / 256, 256, 0, stream>>>(ws, WS_N);

    // item Gram: 250k 4-row chunks over 4096 waves (1024 WGs x 4 waves)
    enmf_gram_wmma_kernel<false><<<1024, 128, 0, stream>>>(item_emb, nullptr,
                                                           n_items, G_item);
    // batch user Gram: 1024 chunks over 256 waves, rows gathered via users
    enmf_gram_wmma_kernel<true><<<64, 128, 0, stream>>>(user_emb, users,
                                                        batch, G_user);

    enmf_pos_kernel<<<batch, 256, 0, stream>>>(users, hist, user_emb, item_emb,
                                               Hw, hist_len, pos_acc);
    enmf_sumsq_kernel<<<512, 256, 0, stream>>>(user_emb,
                                               (long long)in_sizes[2] / 4, unorm2);
    enmf_final_kernel<<<1, 64, 0, stream>>>(G_item, G_user, Hw, pos_acc, unorm2,
                                            (float*)d_out);
}